// ChamferLoss_1099511628354
// MI455X (gfx1250) — compile-verified
//
#include <hip/hip_runtime.h>
#include <hip/hip_bf16.h>

// Chamfer distance on MI455X (gfx1250), wave32.
//
// P[i][j] = ||x_i||^2 + ||y_j||^2 - 2 x_i . y_j,  x = gts (rows), y = preds (cols)
// dist1[j]/idx1[j] = min/argmin over i ; dist2[i]/idx2[i] = min/argmin over j.
//
// Core: V_WMMA_F32_16X16X4_F32 with K=4 used to fold the ||x||^2 term:
//   A_i = (x0, x1, x2, xx) ; B_j = (-2y0, -2y1, -2y2, 1)  =>  D = -2 x.y + xx
// Then P = D + yy_j (per-lane scalar add).

typedef __attribute__((ext_vector_type(2))) float v2f;
typedef __attribute__((ext_vector_type(8))) float v8f;

#define NPTS 20000
#define NTILES 1250          // NPTS / 16
#define WAVES_PER_WG 8
#define WG_THREADS 256       // 8 wave32s

typedef unsigned long long u64;

// Order-preserving float->u32 key (handles negative values from rounding).
__device__ __forceinline__ unsigned fkey(float f) {
    unsigned b = __float_as_uint(f);
    return (b & 0x80000000u) ? ~b : (b | 0x80000000u);
}
__device__ __forceinline__ float funkey(unsigned k) {
    unsigned b = (k & 0x80000000u) ? (k & 0x7FFFFFFFu) : ~k;
    return __uint_as_float(b);
}

// min-pair keeping the earlier (left) element on ties -> first-occurrence argmin
__device__ __forceinline__ void minpair(float va, int ia, float vb, int ib,
                                        float& vo, int& io) {
    bool c = vb < va;
    vo = c ? vb : va;
    io = c ? ib : ia;
}

// ---------------- Kernel 1: precompute B-pack table + init global dist1 -----
// bpack[j] = (-2*y0, -2*y1, -2*y2, yy)
__global__ void __launch_bounds__(256) chamfer_prep(
    const float* __restrict__ preds, float4* __restrict__ bpack,
    u64* __restrict__ gD1) {
    int j = blockIdx.x * blockDim.x + threadIdx.x;
    if (j < NPTS) {
        float y0 = preds[3 * j + 0];
        float y1 = preds[3 * j + 1];
        float y2 = preds[3 * j + 2];
        float yy = fmaf(y0, y0, fmaf(y1, y1, y2 * y2));
        bpack[j] = make_float4(-2.0f * y0, -2.0f * y1, -2.0f * y2, yy);
        gD1[j] = ~0ull;
    }
}

// ---------------- Kernel 2: main WMMA sweep ---------------------------------
// Wave w of block b owns gts rows [strip*16, strip*16+16), strip = b*8+w,
// and sweeps all 1250 column tiles of preds.
// LDS: per-workgroup packed dist1 table (20000 x u64 = 156.25 KB of the WGP's
// 320 KB LDS), merged to global once per workgroup.
__global__ void __launch_bounds__(WG_THREADS) chamfer_main(
    const float* __restrict__ gts, const float4* __restrict__ bpack,
    u64* __restrict__ gD1,
    float* __restrict__ outDist2, int* __restrict__ outIdx2) {
    extern __shared__ u64 ldsD1[];   // [NPTS]

    const int tid = threadIdx.x;
    for (int k = tid; k < NPTS; k += WG_THREADS) ldsD1[k] = ~0ull;
    __syncthreads();

    const int wave = tid >> 5;
    const int lane = tid & 31;
    const int strip = blockIdx.x * WAVES_PER_WG + wave;

    if (strip < NTILES) {            // wave-uniform: EXEC all-ones inside (WMMA req.)
        const int n = lane & 15;     // column-within-tile / row-within-strip (for A)
        const int hi = lane >> 4;    // half-wave select
        const bool lo = (hi == 0);

        // ---- A matrix (16x4 f32): v0 = K0|K2 split low/high lanes, v1 = K1|K3
        const float* xp = gts + (strip * 16 + n) * 3;
        float x0 = xp[0], x1 = xp[1], x2 = xp[2];
        float xx = fmaf(x0, x0, fmaf(x1, x1, x2 * x2));
        v2f a;
        a.x = lo ? x0 : x2;          // K0 (low lanes) / K2 (high lanes)
        a.y = lo ? x1 : xx;          // K1 / K3

        // Running dist2 state: 8 rows per lane (rows rowBase+0..7)
        float bestV[8];
        int bestJ[8];
#pragma unroll
        for (int v = 0; v < 8; ++v) { bestV[v] = __builtin_inff(); bestJ[v] = 0; }
        const int rowBase = strip * 16 + 8 * hi;

        // Software pipeline: prefetch next tile's B-pack while reducing current.
        float4 nb = bpack[n];                    // tile 0
        for (int jt = 0; jt < NTILES; ++jt) {
            const float4 b4 = nb;
            const int jnext = (jt + 1 < NTILES) ? (jt + 1) : (NTILES - 1);
            nb = bpack[jnext * 16 + n];          // L2-resident b128 prefetch

            const int j = jt * 16 + n;
            v2f b;
            b.x = lo ? b4.x : b4.z;              // K0 / K2
            b.y = lo ? b4.y : 1.0f;              // K1 / K3 (pairs with the xx slot)
            v8f c = {};
            // D = A x B : 16x16 tile of (-2 x.y + xx)
            v8f d = __builtin_amdgcn_wmma_f32_16x16x4_f32(
                false, a, false, b, (short)0, c, false, false);

            const float yy = b4.w;
            float p[8];
#pragma unroll
            for (int v = 0; v < 8; ++v) {
                p[v] = d[v] + yy;                // full squared distance
                bool c2 = p[v] < bestV[v];       // dist2 (strict < keeps first idx)
                bestV[v] = c2 ? p[v] : bestV[v];
                bestJ[v] = c2 ? j : bestJ[v];
            }
            // dist1 in-lane row-min: 3-level tree, first-occurrence tie-break
            float m0, m1, m2, m3, m4, m5;
            int i0, i1, i2, i3, i4, i5;
            minpair(p[0], 0, p[1], 1, m0, i0);
            minpair(p[2], 2, p[3], 3, m1, i1);
            minpair(p[4], 4, p[5], 5, m2, i2);
            minpair(p[6], 6, p[7], 7, m3, i3);
            minpair(m0, i0, m1, i1, m4, i4);
            minpair(m2, i2, m3, i3, m5, i5);
            float rv; int ri;
            minpair(m4, i4, m5, i5, rv, ri);

            // Both half-waves atomically min into the same column slot:
            // LDS u64-min is associative, so no cross-lane combine needed and
            // EXEC stays all-ones (no divergence in the WMMA loop).
            u64 cand = ((u64)fkey(rv) << 32) | (unsigned)(rowBase + ri);
            atomicMin(&ldsD1[j], cand);          // DS_MIN_U64, value+argmin in one op
        }

        // ---- dist2 finish: butterfly across the 16 lanes of each half
#pragma unroll
        for (int v = 0; v < 8; ++v) {
            u64 cand = ((u64)fkey(bestV[v]) << 32) | (unsigned)bestJ[v];
#pragma unroll
            for (int mask = 1; mask <= 8; mask <<= 1) {
                u64 oth = __shfl_xor(cand, mask, 32);
                cand = oth < cand ? oth : cand;
            }
            if (n == 0) {
                int row = strip * 16 + 8 * hi + v;
                outDist2[row] = funkey((unsigned)(cand >> 32));
                outIdx2[row] = (int)(cand & 0xFFFFFFFFu);
            }
        }
    }

    // ---- merge this workgroup's dist1 table into the global one
    __syncthreads();
    for (int k = tid; k < NPTS; k += WG_THREADS) {
        atomicMin(&gD1[k], ldsD1[k]);            // GLOBAL_ATOMIC_MIN_U64
    }
}

// ---------------- Kernel 3: unpack global dist1 -----------------------------
__global__ void __launch_bounds__(256) chamfer_finalize(
    const u64* __restrict__ gD1,
    float* __restrict__ outDist1, int* __restrict__ outIdx1) {
    int j = blockIdx.x * blockDim.x + threadIdx.x;
    if (j < NPTS) {
        u64 p = gD1[j];
        outDist1[j] = funkey((unsigned)(p >> 32));
        outIdx1[j] = (int)(p & 0xFFFFFFFFu);
    }
}

extern "C" void kernel_launch(void* const* d_in, const int* in_sizes, int n_in,
                              void* d_out, int out_size, void* d_ws, size_t ws_size,
                              hipStream_t stream) {
    const float* preds = (const float*)d_in[0];   // [20000,3]
    const float* gts   = (const float*)d_in[1];   // [1,20000,3]

    // d_out layout (return order): dist1[f32 x N], dist2[f32 x N],
    //                              idx1[i32 x N],  idx2[i32 x N]
    float* out = (float*)d_out;
    float* outDist1 = out;
    float* outDist2 = out + NPTS;
    int*   outIdx1  = (int*)(out + 2 * NPTS);
    int*   outIdx2  = (int*)(out + 3 * NPTS);

    // Workspace: [0, 160000)  : gD1 (u64 x 20000)
    //            [160000, ...): bpack (float4 x 20000)
    u64*    gD1   = (u64*)d_ws;
    float4* bpack = (float4*)((char*)d_ws + NPTS * sizeof(u64));

    chamfer_prep<<<(NPTS + 255) / 256, 256, 0, stream>>>(preds, bpack, gD1);

    const int nwg = (NTILES + WAVES_PER_WG - 1) / WAVES_PER_WG;  // 157
    chamfer_main<<<nwg, WG_THREADS, NPTS * sizeof(u64), stream>>>(
        gts, bpack, gD1, outDist2, outIdx2);

    chamfer_finalize<<<(NPTS + 255) / 256, 256, 0, stream>>>(gD1, outDist1, outIdx1);
}